// GATBaselineModel_29386166239289
// MI455X (gfx1250) — compile-verified
//
#include <hip/hip_runtime.h>
#include <hip/hip_bf16.h>
#include <math.h>

#define NROI 116
#define HID 128
#define NGRAPH 64
#define NNODES (NGRAPH * NROI)          // 7424
#define EPG (NROI * NROI)               // 13456 edges per graph
#define KSTEPS 29                       // NROI / 4

typedef float v2f __attribute__((ext_vector_type(2)));
typedef float v8f __attribute__((ext_vector_type(8)));

// ---------------------------------------------------------------------------
// 1) LayerNorm -> ReLU over each row of global_adj [116,116]
// ---------------------------------------------------------------------------
__global__ void ln_relu_kernel(const float* __restrict__ ga_in,
                               const float* __restrict__ lg,
                               const float* __restrict__ lb,
                               float* __restrict__ ga_out) {
    __shared__ float r1[128], r2[128];
    const int r = blockIdx.x, t = threadIdx.x;
    float v = (t < NROI) ? ga_in[r * NROI + t] : 0.f;
    r1[t] = v; r2[t] = v * v;
    __syncthreads();
    for (int off = 64; off; off >>= 1) {
        if (t < off) { r1[t] += r1[t + off]; r2[t] += r2[t + off]; }
        __syncthreads();
    }
    const float mu  = r1[0] * (1.f / NROI);
    const float var = r2[0] * (1.f / NROI) - mu * mu;
    if (t < NROI) {
        float y = (v - mu) * rsqrtf(var + 1e-5f) * lg[t] + lb[t];
        ga_out[r * NROI + t] = y > 0.f ? y : 0.f;
    }
}

// ---------------------------------------------------------------------------
// 2) symmetrize + sigmoid -> adj_mask
// ---------------------------------------------------------------------------
__global__ void sym_sigmoid_kernel(const float* __restrict__ ga,
                                   float* __restrict__ mask) {
    const int idx = blockIdx.x * blockDim.x + threadIdx.x;
    if (idx >= EPG) return;
    const int i = idx / NROI, j = idx % NROI;
    const float v = 0.5f * (ga[idx] + ga[j * NROI + i]);
    mask[idx] = 1.f / (1.f + __expf(-v));
}

// ---------------------------------------------------------------------------
// 3) edge-attention constants: ce[h] = dot(We_h, ae_h)
// ---------------------------------------------------------------------------
__global__ void edge_consts_kernel(const float* __restrict__ We1,
                                   const float* __restrict__ ae1,
                                   const float* __restrict__ We2,
                                   const float* __restrict__ ae2,
                                   float* __restrict__ ce) {
    __shared__ float ra[256];
    __shared__ float rb[128];
    const int t = threadIdx.x;
    ra[t] = We1[t] * ae1[t];
    if (t < 128) rb[t] = We2[t] * ae2[t];
    __syncthreads();
    for (int off = 64; off; off >>= 1) {
        if ((t & 127) < off) ra[t] += ra[t + off];
        if (t < off) rb[t] += rb[t + off];
        __syncthreads();
    }
    if (t == 0) { ce[0] = ra[0]; ce[1] = ra[128]; ce[2] = rb[0]; }
}

// ---------------------------------------------------------------------------
// 4) fp32 WMMA GEMM, compile-time K/N: C[M,N] = (A (+ tiled prompt)) @ B[K,N]
//    Block = 256 thr = 8 waves; block tile 128x16. B panel staged in LDS
//    (transposed -> ds_load_b64 fragments), shared by all 8 waves.
// ---------------------------------------------------------------------------
template <int K, int N, bool WITH_PROMPT>
__global__ __launch_bounds__(256) void wmma_gemm_f32(
        const float* __restrict__ A, const float* __restrict__ prompt,
        const float* __restrict__ B, float* __restrict__ C) {
    __shared__ float Bl[16][K];                 // Bl[n][k]
    const int tid  = threadIdx.x;
    const int n0   = blockIdx.x * 16;

    // cooperative transpose-load of the B panel [K,16] -> LDS
    for (int idx = tid; idx < 16 * K; idx += 256) {
        const int k = idx >> 4, n = idx & 15;
        Bl[n][k] = B[k * N + n0 + n];
    }

    const int lane = tid & 31, wave = tid >> 5;
    const int half = lane >> 4, l = lane & 15;
    const int m0   = (blockIdx.y * 8 + wave) * 16;
    const int rowA = m0 + l;

    const float* Ap = A + (size_t)rowA * K + 2 * half;
    const float* Pp = prompt + (rowA % NROI) * K + 2 * half;  // used iff WITH_PROMPT
    __builtin_prefetch(Ap, 0, 0);               // global_prefetch_b8
    __syncthreads();
    const float* Bp = &Bl[l][2 * half];

    v8f acc = {};
#pragma unroll
    for (int k0 = 0; k0 < K; k0 += 4) {
        v2f a, b;
        a.x = Ap[k0];
        a.y = Ap[k0 + 1];
        if (WITH_PROMPT) { a.x += Pp[k0]; a.y += Pp[k0 + 1]; }
        b.x = Bp[k0];
        b.y = Bp[k0 + 1];
        acc = __builtin_amdgcn_wmma_f32_16x16x4_f32(false, a, false, b,
                                                    (short)0, acc, false, false);
    }
#pragma unroll
    for (int v = 0; v < 8; ++v) {               // D: VGPR v -> M = m0 + v + 8*half
        const int row = m0 + v + 8 * half;
        C[(size_t)row * N + n0 + l] = acc[v];
    }
}

// ---------------------------------------------------------------------------
// 5) per-node attention scalars (one wave per (n,h), wave32 shfl reduce)
// ---------------------------------------------------------------------------
template <int H>
__global__ void node_alphas_kernel(const float* __restrict__ hf,
                                   const float* __restrict__ a_src,
                                   const float* __restrict__ a_dst,
                                   float* __restrict__ asn,
                                   float* __restrict__ adn) {
    const int pair = blockIdx.x * 8 + (threadIdx.x >> 5);
    const int lane = threadIdx.x & 31;
    if (pair >= NNODES * H) return;
    const int n = pair / H, h = pair % H;
    const float* hp = hf + (size_t)n * (H * HID) + h * HID;
    const float* as = a_src + h * HID;
    const float* ad = a_dst + h * HID;
    float s = 0.f, d = 0.f;
#pragma unroll
    for (int c0 = 0; c0 < HID; c0 += 32) {
        const float v = hp[c0 + lane];
        s += v * as[c0 + lane];
        d += v * ad[c0 + lane];
    }
#pragma unroll
    for (int off = 16; off; off >>= 1) {
        s += __shfl_xor(s, off, 32);
        d += __shfl_xor(d, off, 32);
    }
    if (lane == 0) { asn[pair] = s; adn[pair] = d; }
}

// ---------------------------------------------------------------------------
// 6) GAT attention + aggregation for one (graph, head):
//    S[s,d] = lrelu(asn[s]+adn[d]+ea*mask*ce) -> column softmax over s
//    out[d,:] = elu( (P^T @ H)[d,:] * inv_sum[d] + bias )
//    B fragments (feature panel) preloaded to registers once; fully unrolled
//    fp32 WMMA chain over the LDS score matrix.
// ---------------------------------------------------------------------------
#define SPITCH 128
template <int H>
__global__ __launch_bounds__(256) void gat_aggregate_kernel(
        const float* __restrict__ hf,
        const float* __restrict__ asn, const float* __restrict__ adn,
        const float* __restrict__ edge_attr, const float* __restrict__ adj_mask,
        const float* __restrict__ ce, const float* __restrict__ bias,
        float* __restrict__ out) {
    constexpr int HS = H * HID;
    __shared__ float S[NROI][SPITCH];           // S[s][d]
    __shared__ float inv_sum[NROI];
    const int gh = blockIdx.x;
    const int g  = gh / H, h = gh % H;
    const int base = g * NROI;
    const int tid  = threadIdx.x;
    const int lane = tid & 31, wave = tid >> 5;
    const int half = lane >> 4, l = lane & 15;
    const int n0   = wave * 16;                 // 8 waves cover 128 channels

    // B-fragment preload: hf[(base+ka)*HS + h*HID + n0 + l], ka = 4*ks + 2*half
    const float* hp = hf + ((size_t)base + 2 * half) * HS + h * HID + n0 + l;
    v2f breg[KSTEPS];
#pragma unroll
    for (int ks = 0; ks < KSTEPS; ++ks) {
        breg[ks].x = hp[(4 * ks) * HS];
        breg[ks].y = hp[(4 * ks + 1) * HS];
    }

    // phase 1: raw attention scores + leaky_relu(0.2)
    const float ceh = ce[h];
    const float* ea = edge_attr + (size_t)g * EPG;
    for (int idx = tid; idx < EPG; idx += 256) {
        const int s = idx / NROI, d = idx % NROI;
        float al = asn[(base + s) * H + h] + adn[(base + d) * H + h]
                 + ea[idx] * adj_mask[idx] * ceh;
        S[s][d] = al > 0.f ? al : 0.2f * al;
    }
    __syncthreads();

    // phase 2: softmax over s for each destination column d
    for (int d = tid; d < NROI; d += 256) {
        float m = -1e30f;
        for (int s = 0; s < NROI; ++s) m = fmaxf(m, S[s][d]);
        float sum = 0.f;
        for (int s = 0; s < NROI; ++s) {
            const float e = __expf(S[s][d] - m);
            S[s][d] = e;
            sum += e;
        }
        inv_sum[d] = 1.f / (sum + 1e-16f);
    }
    __syncthreads();

    // phase 3: out = P^T @ H via fp32 WMMA; normalization folded into epilogue
    const float bterm = bias[h * HID + n0 + l];
#pragma unroll
    for (int mt = 0; mt < 8; ++mt) {            // d tiles 0..127 (>=116 dropped)
        const int d0 = mt * 16;
        v8f acc = {};
#pragma unroll
        for (int ks = 0; ks < KSTEPS; ++ks) {
            const int ka = 4 * ks + 2 * half;
            v2f a;
            a.x = S[ka][d0 + l];                // A[d][s] = P[s][d]
            a.y = S[ka + 1][d0 + l];
            acc = __builtin_amdgcn_wmma_f32_16x16x4_f32(false, a, false, breg[ks],
                                                        (short)0, acc, false, false);
        }
#pragma unroll
        for (int v = 0; v < 8; ++v) {
            const int d = d0 + v + 8 * half;
            if (d < NROI) {
                const float val = acc[v] * inv_sum[d] + bterm;
                out[(size_t)(base + d) * HS + h * HID + n0 + l] =
                    val > 0.f ? val : expm1f(val);   // ELU fused
            }
        }
    }
}

// ---------------------------------------------------------------------------
// 7) global attention pooling + MLP classifier, one block (128 thr) per graph
// ---------------------------------------------------------------------------
__global__ void pool_classify_kernel(const float* __restrict__ h2,
                                     const float* __restrict__ Wg, const float* __restrict__ bg,
                                     const float* __restrict__ Wf, const float* __restrict__ bf,
                                     const float* __restrict__ Wc, const float* __restrict__ bc,
                                     float* __restrict__ out) {
    __shared__ float gate[NROI];
    __shared__ float pooled[HID];
    __shared__ float feat[64];
    __shared__ float invs;
    const int g = blockIdx.x, t = threadIdx.x;
    const float* hb = h2 + (size_t)g * NROI * HID;

    if (t < NROI) {
        float s = bg[0];
        for (int c = 0; c < HID; ++c) s += hb[t * HID + c] * Wg[c];
        gate[t] = s;
    }
    __syncthreads();
    if (t == 0) {
        float m = -1e30f;
        for (int n = 0; n < NROI; ++n) m = fmaxf(m, gate[n]);
        float sum = 0.f;
        for (int n = 0; n < NROI; ++n) {
            const float e = __expf(gate[n] - m);
            gate[n] = e;
            sum += e;
        }
        invs = 1.f / (sum + 1e-16f);
    }
    __syncthreads();
    {
        float p = 0.f;
        for (int n = 0; n < NROI; ++n) p += gate[n] * hb[n * HID + t];
        pooled[t] = p * invs;
    }
    __syncthreads();
    if (t < 64) {
        float f = bf[t];
        for (int c = 0; c < HID; ++c) f += pooled[c] * Wf[c * 64 + t];
        feat[t] = f > 0.f ? f : 0.f;
    }
    __syncthreads();
    if (t < 2) {
        float z = bc[t];
        for (int j = 0; j < 64; ++j) z += feat[j] * Wc[j * 2 + t];
        out[g * 2 + t] = z;
    }
}

// ---------------------------------------------------------------------------
extern "C" void kernel_launch(void* const* d_in, const int* in_sizes, int n_in,
                              void* d_out, int out_size, void* d_ws, size_t ws_size,
                              hipStream_t stream) {
    const float* x          = (const float*)d_in[0];
    const float* edge_attr  = (const float*)d_in[1];
    const float* softp      = (const float*)d_in[2];
    const float* global_adj = (const float*)d_in[3];
    const float* ln_g       = (const float*)d_in[4];
    const float* ln_b       = (const float*)d_in[5];
    const float* W1  = (const float*)d_in[6];
    const float* as1 = (const float*)d_in[7];
    const float* ad1 = (const float*)d_in[8];
    const float* We1 = (const float*)d_in[9];
    const float* ae1 = (const float*)d_in[10];
    const float* b1  = (const float*)d_in[11];
    const float* W2  = (const float*)d_in[12];
    const float* as2 = (const float*)d_in[13];
    const float* ad2 = (const float*)d_in[14];
    const float* We2 = (const float*)d_in[15];
    const float* ae2 = (const float*)d_in[16];
    const float* b2  = (const float*)d_in[17];
    const float* Wg  = (const float*)d_in[18];
    const float* bg  = (const float*)d_in[19];
    const float* Wf  = (const float*)d_in[20];
    const float* bf  = (const float*)d_in[21];
    const float* Wc  = (const float*)d_in[22];
    const float* bc  = (const float*)d_in[23];
    // edge_index / batch unused: graph structure is static dense per-graph.

    float* ws    = (float*)d_ws;
    float* ga    = ws;                       // 13456
    float* mask  = ga    + 13456;            // 13456
    float* ce    = mask  + 13456;            // 4
    float* h1    = ce    + 4;                // 7424*256
    float* asn1  = h1    + NNODES * 256;     // 7424*2
    float* adn1  = asn1  + NNODES * 2;       // 7424*2
    float* out1  = adn1  + NNODES * 2;       // 7424*256 (post-ELU)
    float* h2lin = out1  + NNODES * 256;     // 7424*128
    float* asn2  = h2lin + NNODES * HID;     // 7424
    float* adn2  = asn2  + NNODES;           // 7424
    float* h2    = adn2  + NNODES;           // 7424*128 (post-ELU)

    // learnable adjacency mask
    ln_relu_kernel<<<NROI, 128, 0, stream>>>(global_adj, ln_g, ln_b, ga);
    sym_sigmoid_kernel<<<(EPG + 255) / 256, 256, 0, stream>>>(ga, mask);
    edge_consts_kernel<<<1, 256, 0, stream>>>(We1, ae1, We2, ae2, ce);

    // conv1: h1 = (x + prompt) @ W1   [7424,116]@[116,256]
    wmma_gemm_f32<NROI, 256, true><<<dim3(256 / 16, NNODES / 128), 256, 0, stream>>>(
        x, softp, W1, h1);
    node_alphas_kernel<2><<<(NNODES * 2) / 8, 256, 0, stream>>>(
        h1, as1, ad1, asn1, adn1);
    gat_aggregate_kernel<2><<<NGRAPH * 2, 256, 0, stream>>>(
        h1, asn1, adn1, edge_attr, mask, ce, b1, out1);

    // conv2: h2lin = elu(out1) @ W2   [7424,256]@[256,128]
    wmma_gemm_f32<256, HID, false><<<dim3(HID / 16, NNODES / 128), 256, 0, stream>>>(
        out1, nullptr, W2, h2lin);
    node_alphas_kernel<1><<<NNODES / 8, 256, 0, stream>>>(
        h2lin, as2, ad2, asn2, adn2);
    gat_aggregate_kernel<1><<<NGRAPH, 256, 0, stream>>>(
        h2lin, asn2, adn2, edge_attr, mask, ce + 2, b2, h2);

    // pooling + classifier -> logits [64,2]
    pool_classify_kernel<<<NGRAPH, 128, 0, stream>>>(
        h2, Wg, bg, Wf, bf, Wc, bc, (float*)d_out);
}